// MaskedAttention_20572893348312
// MI455X (gfx1250) — compile-verified
//
#include <hip/hip_runtime.h>
#include <stdint.h>

// Top-k row masking (k = 64, N = 8192): keep each row's 64 largest values,
// zero the rest. One 256-thread workgroup (8 wave32) per row.
//
// CDNA5 path: row staged in LDS via GLOBAL_LOAD_ASYNC_TO_LDS_B128 (ASYNCcnt),
// radix-select threshold via per-wave LDS-atomic histograms + parallel suffix
// scan, masked row streamed back with GLOBAL_STORE_ASYNC_FROM_LDS_B128.

namespace {

constexpr int kNCols        = 8192;
constexpr int kK            = 64;
constexpr int kBlock        = 256;
constexpr int kWaves        = kBlock / 32;          // 8 wave32 per block
constexpr int kVecPerThread = kNCols / 4 / kBlock;  // 8 float4 chunks per thread

// Order-preserving float -> uint key (larger float => larger key).
__device__ __forceinline__ unsigned f2key(float f) {
  unsigned u = __float_as_uint(f);
  return (u & 0x80000000u) ? ~u : (u | 0x80000000u);
}

// INST_OFFSET is added to BOTH the LDS and global addresses (ISA 08, §4.4),
// and our LDS/global chunks advance in lockstep -> one reg pair, 8 offsets.
#define ASYNC_LOAD_CHUNK(OFF)                                        \
  asm volatile("global_load_async_to_lds_b128 %0, %1, off offset:" #OFF \
               :: "v"(laddr0), "v"(gaddr_in) : "memory")
#define ASYNC_STORE_CHUNK(OFF)                                        \
  asm volatile("global_store_async_from_lds_b128 %0, %1, off offset:" #OFF \
               :: "v"(gaddr_out), "v"(laddr0) : "memory")

__global__ __launch_bounds__(kBlock) void topk_mask_kernel(
    const float* __restrict__ X, float* __restrict__ Out) {
  __shared__ __align__(16) float row[kNCols];     // 32 KB staged row
  __shared__ unsigned hist[kWaves][256];          // per-wave histograms (8 KB)
  __shared__ unsigned scan[256];                  // suffix-sum workspace
  __shared__ unsigned sel_val;    // decided high bits of the k-th largest key
  __shared__ unsigned sel_mask;   // mask of decided bits
  __shared__ unsigned sel_remk;   // remaining k within the decided prefix
  __shared__ int      eq_budget;  // how many threshold-ties to keep

  const int t    = threadIdx.x;
  const int wave = t >> 5;
  const uint64_t row_bytes = (uint64_t)kNCols * sizeof(float);
  const uint32_t lane_byte = (uint32_t)t * 16u;
  const uint64_t gaddr_in  =
      (uint64_t)(uintptr_t)X + (uint64_t)blockIdx.x * row_bytes + lane_byte;
  const uint64_t gaddr_out =
      (uint64_t)(uintptr_t)Out + (uint64_t)blockIdx.x * row_bytes + lane_byte;
  // LDS byte offset of `row`: flat shared-aperture address truncated to 32 bits.
  const uint32_t laddr0 = (uint32_t)(uintptr_t)&row[0] + lane_byte;

  // ---- 1) async global -> LDS copy of the row (8 x b128 per lane) ----
  ASYNC_LOAD_CHUNK(0);
  ASYNC_LOAD_CHUNK(4096);
  ASYNC_LOAD_CHUNK(8192);
  ASYNC_LOAD_CHUNK(12288);
  ASYNC_LOAD_CHUNK(16384);
  ASYNC_LOAD_CHUNK(20480);
  ASYNC_LOAD_CHUNK(24576);
  ASYNC_LOAD_CHUNK(28672);
  if (t == 0) { sel_val = 0u; sel_mask = 0u; sel_remk = kK; eq_budget = 0; }
  asm volatile("s_wait_asynccnt 0" ::: "memory");
  __syncthreads();

  // ---- 2) MSB-first 8-bit radix select of the k-th largest key ----
  for (int shift = 24; shift >= 0; shift -= 8) {
    // zero all per-wave histograms (2048 words / 256 threads)
#pragma unroll
    for (int i = 0; i < kWaves; ++i) (&hist[0][0])[t + i * kBlock] = 0u;
    __syncthreads();

    const unsigned pv = sel_val, pm = sel_mask;
    unsigned* myhist = hist[wave];
#pragma unroll
    for (int j = 0; j < kVecPerThread; ++j) {
      const int base  = (t + j * kBlock) * 4;
      const float4 v  = *(const float4*)&row[base];  // ds_load_b128
      const unsigned k0 = f2key(v.x), k1 = f2key(v.y),
                     k2 = f2key(v.z), k3 = f2key(v.w);
      if ((k0 & pm) == pv) atomicAdd(&myhist[(k0 >> shift) & 255u], 1u);
      if ((k1 & pm) == pv) atomicAdd(&myhist[(k1 >> shift) & 255u], 1u);
      if ((k2 & pm) == pv) atomicAdd(&myhist[(k2 >> shift) & 255u], 1u);
      if ((k3 & pm) == pv) atomicAdd(&myhist[(k3 >> shift) & 255u], 1u);
    }
    __syncthreads();

    // Reduce per-wave copies: thread t owns bin t.
    unsigned total = 0;
#pragma unroll
    for (int w = 0; w < kWaves; ++w) total += hist[w][t];
    scan[t] = total;
    __syncthreads();

    // Inclusive suffix scan (Hillis-Steele, 8 steps).
    unsigned v = total;
#pragma unroll
    for (int off = 1; off < 256; off <<= 1) {
      const unsigned add = (t + off < 256) ? scan[t + off] : 0u;
      __syncthreads();
      v += add;
      scan[t] = v;
      __syncthreads();
    }

    const unsigned remk  = sel_remk;
    const unsigned s     = scan[t];
    const unsigned snext = (t < 255) ? scan[t + 1] : 0u;
    __syncthreads();  // everyone has read sel_remk/scan before winner writes
    if (s >= remk && snext < remk) {  // unique winner: digit = t
      sel_val  = pv | ((unsigned)t << shift);
      sel_mask = pm | (0xFFu << shift);
      sel_remk = remk - snext;
      if (shift == 0) eq_budget = (int)(remk - snext);
    }
    __syncthreads();
  }

  const unsigned thr = sel_val;  // full 32-bit key of the 64th-largest element

  // ---- 3) mask in place in LDS: keep key > thr, plus eq_budget ties ----
#pragma unroll
  for (int j = 0; j < kVecPerThread; ++j) {
    const int base = (t + j * kBlock) * 4;
    float4 v = *(const float4*)&row[base];
    float* p = reinterpret_cast<float*>(&v);
#pragma unroll
    for (int e = 0; e < 4; ++e) {
      const unsigned k = f2key(p[e]);
      bool keep = (k > thr);
      if (k == thr) keep = (atomicSub(&eq_budget, 1) > 0);
      if (!keep) p[e] = 0.0f;
    }
    *(float4*)&row[base] = v;  // ds_store_b128
  }

  // This wave's LDS writes must land before the async engine reads them.
  asm volatile("s_wait_dscnt 0" ::: "memory");

  // ---- 4) async LDS -> global store of the masked row ----
  ASYNC_STORE_CHUNK(0);
  ASYNC_STORE_CHUNK(4096);
  ASYNC_STORE_CHUNK(8192);
  ASYNC_STORE_CHUNK(12288);
  ASYNC_STORE_CHUNK(16384);
  ASYNC_STORE_CHUNK(20480);
  ASYNC_STORE_CHUNK(24576);
  ASYNC_STORE_CHUNK(28672);
  // S_ENDPGM performs an implicit wait-idle (ASYNCcnt drains before exit).
}

}  // namespace

extern "C" void kernel_launch(void* const* d_in, const int* in_sizes, int n_in,
                              void* d_out, int out_size, void* d_ws, size_t ws_size,
                              hipStream_t stream) {
  (void)n_in; (void)d_ws; (void)ws_size; (void)out_size;
  const float* X = (const float*)d_in[0];
  float* Out     = (float*)d_out;
  // pooling_size (d_in[1]) is fixed at 64 per the reference setup.
  const int B = in_sizes[0] / kNCols;  // 4096 rows
  topk_mask_kernel<<<dim3(B), dim3(kBlock), 0, stream>>>(X, Out);
}